// NeuralNetwork_42975442764252
// MI455X (gfx1250) — compile-verified
//
#include <hip/hip_runtime.h>

typedef __attribute__((ext_vector_type(2))) float v2f;
typedef __attribute__((ext_vector_type(8))) float v8f;

#define LENGTH   6783
#define T_STRIDE 755
#define NBANDS   9
#define OUT_BASE (LENGTH - T_STRIDE)   /* 6028 */
#define NTILES   48                    /* ceil(755/16) */
#define NWAVES   8
#define TILES_PER_WAVE (NTILES / NWAVES)   /* 6, exact */

// Branchless masked gather: clamp to a legal address (unconditional
// global_load_b32), then v_cndmask the out-of-range lanes to zero.
// Never touches EXEC, so the WMMA all-1s EXEC invariant holds trivially.
__device__ __forceinline__ float q_masked(const float* __restrict__ q, int idx) {
  const bool valid = ((unsigned)idx < (unsigned)LENGTH);
  int cidx = idx < 0 ? 0 : idx;
  cidx = cidx > (LENGTH - 1) ? (LENGTH - 1) : cidx;
  const float v = q[cidx];
  return valid ? v : 0.0f;
}

// Single-workgroup kernel: evaluates the 9-band Toeplitz matvec on the last
// T outputs via f32 WMMA (16x16x4), squares, and block-reduces to one scalar.
__global__ __launch_bounds__(256) void banded_toeplitz_loss(
    const float* __restrict__ q,
    const float* __restrict__ alpha_p,
    float* __restrict__ out) {
  const int tid  = threadIdx.x;
  const int wave = tid >> 5;          // 8 waves of 32 (wave32)
  const int lane = tid & 31;
  const int col  = lane & 15;         // N index within a 16-wide tile
  const bool hi  = (lane >= 16);      // upper half-wave holds K+2 / M+8 slots

  // Band coefficients c_i = (-alpha)^i, straight-line (no register-array indexing).
  const float m = -alpha_p[0];
  const float c0 = 1.0f;
  const float c1 = m;
  const float c2 = c1 * m;
  const float c3 = c2 * m;
  const float c4 = c3 * m;
  const float c5 = c4 * m;
  const float c6 = c5 * m;
  const float c7 = c6 * m;
  const float c8 = c7 * m;

  // A-matrix operands per K-pass (tile-invariant; 6 cndmasks total).
  // ISA layout, f32 16x16x4 A: VGPR0 = K {4p | 4p+2} (lanes 0-15 | 16-31),
  // VGPR1 = K {4p+1 | 4p+3}; all 16 rows equal c_k.
  v2f a0, a1, a2;
  a0.x = hi ? c2 : c0;   a0.y = hi ? c3 : c1;    // K = 0..3
  a1.x = hi ? c6 : c4;   a1.y = hi ? c7 : c5;    // K = 4..7
  a2.x = hi ? 0.0f : c8; a2.y = 0.0f;            // K = 8..11 (c9..c11 = 0)

  // Per-pass K bases for this lane's B slots (B mirrors A's K mapping, N = lane&15).
  const int kb = hi ? 2 : 0;

  float s = 0.0f;
  // Compile-time trip count: no EXEC-masked loop machinery, EXEC stays all-1s.
  for (int k = 0; k < TILES_PER_WAVE; ++k) {
    const int t  = wave + NWAVES * k;
    const int j  = t * 16 + col;      // output column (0..767; >=755 is padding)

    v2f b0, b1, b2;
    // B[K][j] = q[OUT_BASE + j - K*T], zero outside the tensor.
    b0.x = q_masked(q, OUT_BASE + j - (kb + 0) * T_STRIDE);
    b0.y = q_masked(q, OUT_BASE + j - (kb + 1) * T_STRIDE);
    b1.x = q_masked(q, OUT_BASE + j - (kb + 4) * T_STRIDE);
    b1.y = q_masked(q, OUT_BASE + j - (kb + 5) * T_STRIDE);
    b2.x = q_masked(q, OUT_BASE + j - (kb + 8) * T_STRIDE); // K=8 (lo) / self-masks (hi, K=10)
    b2.y = 0.0f;                                            // K=9/11: zero coefficient

    // D = A*B + C, three accumulating v_wmma_f32_16x16x4_f32
    v8f acc = {};
    acc = __builtin_amdgcn_wmma_f32_16x16x4_f32(false, a0, false, b0, (short)0, acc, false, false);
    acc = __builtin_amdgcn_wmma_f32_16x16x4_f32(false, a1, false, b1, (short)0, acc, false, false);
    acc = __builtin_amdgcn_wmma_f32_16x16x4_f32(false, a2, false, b2, (short)0, acc, false, false);

    // All D rows are identical; VGPR0: lane L holds y_{16t + (L&15)} (duplicated in hi half).
    const float y = acc[0];
    s += (!hi && j < T_STRIDE) ? y * y : 0.0f;  // count each column once, mask padding
  }

  // Block tree-reduction in LDS; one scalar store.
  __shared__ float red[256];
  red[tid] = s;
  __syncthreads();
  #pragma unroll
  for (int stride = 128; stride > 0; stride >>= 1) {
    if (tid < stride) red[tid] += red[tid + stride];
    __syncthreads();
  }
  if (tid == 0) out[0] = red[0];
}

extern "C" void kernel_launch(void* const* d_in, const int* in_sizes, int n_in,
                              void* d_out, int out_size, void* d_ws, size_t ws_size,
                              hipStream_t stream) {
  (void)in_sizes; (void)n_in; (void)out_size; (void)d_ws; (void)ws_size;
  const float* q     = (const float*)d_in[0];  // [1,1,6783] f32
  const float* alpha = (const float*)d_in[1];  // scalar f32
  float* out = (float*)d_out;                  // scalar loss
  banded_toeplitz_loss<<<1, 256, 0, stream>>>(q, alpha, out);
}